// SparseFullyConnectedLayer_35424890258184
// MI455X (gfx1250) — compile-verified
//
#include <hip/hip_runtime.h>
#include <math.h>

#define B_ROWS 4096
#define O_DIM  512

typedef __attribute__((ext_vector_type(2))) float v2f;
typedef __attribute__((ext_vector_type(8))) float v8f;

struct Meta {
    int   r[4];
    int   c[4];
    float v[4];
};

__device__ __forceinline__ int lower_bound_i32(const int* __restrict__ arr, int n, int key) {
    int lo = 0, hi = n;
    while (lo < hi) {
        int mid = (lo + hi) >> 1;
        if (arr[mid] < key) lo = mid + 1; else hi = mid;
    }
    return lo;
}

// All addresses uniform (k is an SGPR) -> scalar broadcast loads (s_load_b128 x3).
__device__ __forceinline__ Meta load_meta(const int* __restrict__ rows,
                                          const int* __restrict__ cols,
                                          const float* __restrict__ vals, int k) {
    Meta m;
#pragma unroll
    for (int j = 0; j < 4; ++j) {
        m.r[j] = rows[k + j];
        m.c[j] = cols[k + j];
        m.v[j] = vals[k + j];
    }
    return m;
}

// Tail chunk: clamp indices (k < ke guaranteed); invalid slots get val=0 (=> A=0) and col=0 (safe addr).
__device__ __forceinline__ Meta load_meta_guard(const int* __restrict__ rows,
                                                const int* __restrict__ cols,
                                                const float* __restrict__ vals, int k, int ke) {
    Meta m;
#pragma unroll
    for (int j = 0; j < 4; ++j) {
        const bool ok = (k + j) < ke;       // uniform
        const int  i  = ok ? (k + j) : k;   // uniform, in-bounds
        m.r[j] = rows[i];
        m.c[j] = ok ? cols[i] : 0;
        m.v[j] = ok ? vals[i] : 0.f;
    }
    return m;
}

__device__ __forceinline__ float fast_tanh(float x) {
    const float ax = fabsf(x);
    const float t  = __expf(-2.0f * ax);                 // t in (0,1], no overflow
    const float y  = __fdividef(1.0f - t, 1.0f + t);
    return copysignf(y, x);
}

// One block = 16 batch rows; 8 waves; each wave owns 4 column tiles (64 of 512 cols).
// C = A x B with A[m,k] = (rows[k]==r0+m) ? vals[k] : 0, B[k,n] = W[cols[k],n] (exact fp32).
__global__ __launch_bounds__(256) void spmm_wmma_f32_kernel(
    const int*   __restrict__ rows,
    const int*   __restrict__ cols,
    const float* __restrict__ vals,
    const float* __restrict__ W,
    const float* __restrict__ bias,
    float*       __restrict__ out,
    int nnz)
{
    const int r0   = blockIdx.x * 16;
    const int lane = threadIdx.x & 31;
    const int wave = threadIdx.x >> 5;                  // 0..7

    const int ks = lower_bound_i32(rows, nnz, r0);
    const int ke = lower_bound_i32(rows, nnz, r0 + 16);

    const bool     hi    = lane >= 16;                  // lanes 16-31 hold K slots {2,3}
    const int      m     = lane & 15;
    const int      r0m   = r0 + m;
    const unsigned nbase = (unsigned)(wave * 64 + m);   // contiguous 256B W segment per wave/chunk

    v8f acc0 = {}, acc1 = {}, acc2 = {}, acc3 = {};

    auto process = [&](const Meta& mt) {
        const int   rA = hi ? mt.r[2] : mt.r[0];
        const int   rB = hi ? mt.r[3] : mt.r[1];
        const int   cA = hi ? mt.c[2] : mt.c[0];
        const int   cB = hi ? mt.c[3] : mt.c[1];
        const float vA = hi ? mt.v[2] : mt.v[0];
        const float vB = hi ? mt.v[3] : mt.v[1];

        v2f A;
        A.x = (rA == r0m) ? vA : 0.f;
        A.y = (rB == r0m) ? vB : 0.f;

        // 32-bit element offsets (< 2^26) -> GVS addressing: global_load v, voff, s[W] scale_offset
        const unsigned off0 = (unsigned)cA * O_DIM + nbase;
        const unsigned off1 = (unsigned)cB * O_DIM + nbase;

        v2f Bm;
        Bm.x = W[off0 +  0]; Bm.y = W[off1 +  0];
        acc0 = __builtin_amdgcn_wmma_f32_16x16x4_f32(false, A, false, Bm, (short)0, acc0, false, false);
        Bm.x = W[off0 + 16]; Bm.y = W[off1 + 16];
        acc1 = __builtin_amdgcn_wmma_f32_16x16x4_f32(false, A, false, Bm, (short)0, acc1, false, false);
        Bm.x = W[off0 + 32]; Bm.y = W[off1 + 32];
        acc2 = __builtin_amdgcn_wmma_f32_16x16x4_f32(false, A, false, Bm, (short)0, acc2, false, false);
        Bm.x = W[off0 + 48]; Bm.y = W[off1 + 48];
        acc3 = __builtin_amdgcn_wmma_f32_16x16x4_f32(false, A, false, Bm, (short)0, acc3, false, false);
    };

    const int nfull = (ke - ks) >> 2;                   // full 4-nnz chunks
    int k = ks;

    if (nfull > 0) {
        // depth-2 metadata pipeline + W prefetch two chunks ahead
        Meta m0 = load_meta(rows, cols, vals, k);
        Meta m1 = (nfull > 1) ? load_meta(rows, cols, vals, k + 4) : m0;
        for (int t = 0; t < nfull; ++t, k += 4) {
            Meta m2 = m1;
            if (t + 2 < nfull) {
                m2 = load_meta(rows, cols, vals, k + 8);
                const unsigned pA = (unsigned)(hi ? m2.c[2] : m2.c[0]) * O_DIM + nbase;
                const unsigned pB = (unsigned)(hi ? m2.c[3] : m2.c[1]) * O_DIM + nbase;
                __builtin_prefetch(W + pA, 0, 0);       // loc=0: aim for WGP-scope (all cache levels)
                __builtin_prefetch(W + pB, 0, 0);
            }
            process(m0);
            m0 = m1;
            m1 = m2;
        }
    }
    if (k < ke) {                                       // tail: at most one guarded chunk
        process(load_meta_guard(rows, cols, vals, k, ke));
    }

    // ---- epilogue: tanh(acc + b[n]); C layout: VGPR j -> row r0 + j + 8*(lane>=16) ----
    const int   rowOfs = r0 + (hi ? 8 : 0);
    const float b0 = bias[nbase +  0];
    const float b1 = bias[nbase + 16];
    const float b2 = bias[nbase + 32];
    const float b3 = bias[nbase + 48];

#pragma unroll
    for (int j = 0; j < 8; ++j) {
        const unsigned base = (unsigned)(rowOfs + j) * O_DIM + nbase;
        out[base +  0] = fast_tanh(acc0[j] + b0);
        out[base + 16] = fast_tanh(acc1[j] + b1);
        out[base + 32] = fast_tanh(acc2[j] + b2);
        out[base + 48] = fast_tanh(acc3[j] + b3);
    }
}

extern "C" void kernel_launch(void* const* d_in, const int* in_sizes, int n_in,
                              void* d_out, int out_size, void* d_ws, size_t ws_size,
                              hipStream_t stream) {
    const int*   rows = (const int*)d_in[0];
    const int*   cols = (const int*)d_in[1];
    const float* vals = (const float*)d_in[2];
    const float* W    = (const float*)d_in[3];
    const float* bias = (const float*)d_in[4];
    float*       out  = (float*)d_out;
    const int nnz = in_sizes[0];

    dim3 grid(B_ROWS / 16);
    dim3 block(256);
    hipLaunchKernelGGL(spmm_wmma_f32_kernel, grid, block, 0, stream,
                       rows, cols, vals, W, bias, out, nnz);
}